// ZeroUpsampling_5978594476536
// MI455X (gfx1250) — compile-verified
//
#include <hip/hip_runtime.h>

// ZeroUpsampling (2,2): out[b,c,2i,2j] = x[b,c,i,j], zeros elsewhere.
// Pure streaming scatter: 265 MB read + 1.06 GB written -> HBM-bound,
// floor ~57us @ 23.3 TB/s. Strategy: one pass, B128 non-temporal stores,
// fully contiguous per-wave address streams.

typedef float __attribute__((ext_vector_type(2))) f2;
typedef float __attribute__((ext_vector_type(4))) f4;

namespace zup {
constexpr int B = 4, C = 32, H = 540, W = 960;
constexpr int OH = H * 2;          // 1080
constexpr int OW = W * 2;          // 1920
constexpr unsigned W2 = W / 2;     // 480 float2-units per input row
constexpr unsigned N  = (unsigned)B * C * H * W2;  // 33,177,600 threads
}

__global__ __launch_bounds__(256)
void ZeroUpsampling_5978594476536_kernel(const float* __restrict__ x,
                                         float* __restrict__ out) {
    using namespace zup;
    unsigned idx = blockIdx.x * blockDim.x + threadIdx.x;
    if (idx >= N) return;

    // idx -> (bc, row, w2); divisors are compile-time constants -> mul/shift
    unsigned w2  = idx % W2;
    unsigned t   = idx / W2;        // = bc*H + row
    unsigned row = t % H;
    unsigned bc  = t / H;           // = b*C + c (fused, layout is contiguous)

    // Load 2 consecutive input floats (8B per lane, contiguous across wave32)
    const f2* src = (const f2*)(x + (size_t)bc * ((size_t)H * W)
                                  + (size_t)row * W) + w2;
    f2 v = __builtin_nontemporal_load(src);

    // Even output row: {x0, 0, x1, 0}; odd output row: all zeros.
    f4 even = { v.x, 0.0f, v.y, 0.0f };
    f4 zero = { 0.0f, 0.0f, 0.0f, 0.0f };

    float* obase = out + (size_t)bc * ((size_t)OH * OW)
                       + (size_t)(2u * row) * OW
                       + (size_t)w2 * 4u;

    // 16B stores, 16B lane stride -> each store instruction writes a
    // contiguous 512B span per wave. NT hint: streamed once, bypass retention.
    __builtin_nontemporal_store(even, (f4*)obase);
    __builtin_nontemporal_store(zero, (f4*)(obase + OW));
}

extern "C" void kernel_launch(void* const* d_in, const int* in_sizes, int n_in,
                              void* d_out, int out_size, void* d_ws, size_t ws_size,
                              hipStream_t stream) {
    const float* x = (const float*)d_in[0];
    float* out = (float*)d_out;

    constexpr unsigned threads = 256;
    constexpr unsigned blocks  = (zup::N + threads - 1) / threads;  // 129,600
    ZeroUpsampling_5978594476536_kernel<<<blocks, threads, 0, stream>>>(x, out);
}